// RenderingModel_50216757625363
// MI455X (gfx1250) — compile-verified
//
#include <hip/hip_runtime.h>
#include <hip/hip_bf16.h>
#include <stdint.h>

// Problem constants (match reference)
#define HH 512
#define WW 512
#define NSPLAT 8192
#define NTIL 64            // 8x8 tiles of 64x64
#define CHUNK_SPLATS 2048  // splats staged into LDS per TDM load
#define LDS_INTS (CHUNK_SPLATS * 3)

#if defined(__HIP_DEVICE_COMPILE__)
#define RFL(x)    __builtin_amdgcn_readfirstlane(x)
#define RDL(v, l) __builtin_amdgcn_readlane((v), (l))
#else
#define RFL(x)    (x)
#define RDL(v, l) (v)
#endif

#if defined(__HIP_DEVICE_COMPILE__) && defined(__gfx1250__) && \
    __has_builtin(__builtin_amdgcn_tensor_load_to_lds)
#define HAVE_TDM 1
typedef unsigned int u32x4 __attribute__((ext_vector_type(4)));
typedef int          i32x4 __attribute__((ext_vector_type(4)));
typedef int          i32x8 __attribute__((ext_vector_type(8)));
#endif

// One block = (tile, chunk). Accumulates the 64x64 tile (16 f32 regs/thread)
// over splats [chunk*perChunk, chunk*perChunk+perChunk), staged through LDS
// by the Tensor Data Mover in 2048-splat sub-chunks.
__global__ __launch_bounds__(256) void splat_gather_accum(
    const int* __restrict__ phw, const float* __restrict__ filt,
    float* __restrict__ dst, int total, int perChunk)
{
  __shared__ int sPhw[LDS_INTS];

  const int b      = blockIdx.x;
  const int tile   = b & (NTIL - 1);
  const int chunk  = b >> 6;               // b / NTIL
  const int tileY0 = (tile >> 3) << 6;
  const int tileX0 = (tile & 7) << 6;
  const int tid    = threadIdx.x;
  const int lane   = tid & 31;
  const int x      = tid & 63;             // column within tile
  // Row-group base: constant within a wave32 -> force it scalar so the
  // per-slot fy bounds checks below become SALU branches, not exec masking.
  const int yb     = RFL((tid >> 6) << 4); // this thread owns rows yb..yb+15
  const int gx     = tileX0 + x;

  float acc[16];
#pragma unroll
  for (int i = 0; i < 16; ++i) acc[i] = 0.0f;

  const int begin = chunk * perChunk;
  const int endS  = min(begin + perChunk, total);

  for (int base = begin; base < endS; base += CHUNK_SPLATS) {
    const int cnt   = min(CHUNK_SPLATS, endS - base);  // always multiple of 32 here
    const int nInts = cnt * 3;

#if defined(HAVE_TDM)
    if (tid < 32) {  // wave 0 issues one DMA for the whole chunk
      const uint64_t ga     = (uint64_t)(uintptr_t)(phw + (size_t)base * 3);
      const uint32_t ldsOff = (uint32_t)(uintptr_t)(&sPhw[0]);  // flat low 32b = LDS offset
      u32x4 g0;
      g0[0] = 1u;                                   // count=1 valid descriptor
      g0[1] = ldsOff;                               // lds_addr
      g0[2] = (uint32_t)ga;                         // global_addr[31:0]
      g0[3] = ((uint32_t)(ga >> 32) & 0x01FFFFFFu)  // global_addr[56:32]
              | 0x80000000u;                        // type=2 ("image")
      i32x8 g1;
      g1[0] = (int)(2u << 16);                      // data_size=4B; mask/flags=0
      g1[1] = (int)((nInts & 0xFFFF) << 16);        // tensor_dim0[15:0]
      g1[2] = (int)(((unsigned)nInts >> 16) | (1u << 16));  // tensor_dim0[31:16], tensor_dim1=1
      g1[3] = (int)((nInts & 0xFFFF) << 16);        // tensor_dim1 hi=0, tile_dim0=nInts
      g1[4] = 0;                                    // tile_dim1=0 (1D), tile_dim2=0
      g1[5] = nInts;                                // tensor_dim0_stride
      g1[6] = 0;
      g1[7] = 0;
      const i32x4 gz4 = {0, 0, 0, 0};               // groups 2/3 unused (<=2D)
      const i32x8 gz8 = {0, 0, 0, 0, 0, 0, 0, 0};   // extra descriptor group (unused)
      __builtin_amdgcn_tensor_load_to_lds(g0, g1, gz4, gz4, gz8, 0);
      __builtin_amdgcn_s_wait_tensorcnt(0);
    }
#else
    for (int i = tid; i < nInts; i += 256) sPhw[i] = phw[(size_t)base * 3 + i];
#endif
    __syncthreads();

    // Ballot-compacted scan: each lane tests one splat; matches are broadcast.
    const int nGroups = cnt >> 5;
    for (int k = 0; k < nGroups; ++k) {
      const int s  = (k << 5) + lane;
      const int p3 = s * 3;
      const int pv = sPhw[p3 + 0];
      const int rv = sPhw[p3 + 1];
      const int cv = sPhw[p3 + 2];
      const bool ok = ((unsigned)(rv - tileY0 + 31) < 127u) &&
                      ((unsigned)(cv - tileX0 + 31) < 127u);
      unsigned m = (unsigned)__ballot(ok);
      while (m) {
        const int l0 = __builtin_ctz(m);
        m &= m - 1;
        const int pu = RDL(pv, l0);                  // SGPR broadcasts
        const int ru = RDL(rv, l0);
        const int cu = RDL(cv, l0);
        const float* fbase = filt + ((size_t)pu << 12);
        __builtin_prefetch(fbase, 0, 0);             // global_prefetch_b8 into L0/L2
        const int fx = gx - cu + 32;                 // filter column for this thread
        if ((unsigned)fx < 64u) {                    // vector (exec) guard on columns
          const float* fcol   = fbase + fx;
          const int    fyBase = tileY0 + yb - ru + 32;  // wave-uniform (scalar)
#pragma unroll
          for (int i = 0; i < 16; ++i) {
            const int fy = fyBase + i;
            if ((unsigned)fy < 64u)                  // scalar guard: s_cmp + s_cbranch
              acc[i] += fcol[fy << 6];               // coalesced L2-hit loads
          }
        }
      }
    }
    __syncthreads();  // LDS reuse fence before next TDM load
  }

  float* o = dst + (size_t)chunk * (HH * WW);
#pragma unroll
  for (int i = 0; i < 16; ++i) {
    const int gy = tileY0 + yb + i;
    o[(gy << 9) + gx] = acc[i];
  }
}

// Deterministic fixed-order reduction of C partial canvases.
__global__ __launch_bounds__(256) void reduce_partials(
    const float* __restrict__ part, float* __restrict__ out, int C)
{
  const int p  = blockIdx.x * 256 + threadIdx.x;
  const int HW = HH * WW;
  float v = 0.0f;
  for (int c = 0; c < C; ++c) v += part[(size_t)c * HW + p];
  out[p] = v;
}

extern "C" void kernel_launch(void* const* d_in, const int* in_sizes, int n_in,
                              void* d_out, int out_size, void* d_ws, size_t ws_size,
                              hipStream_t stream) {
  const int*   phw  = (const int*)d_in[0];    // [8192, 3] int32 (part,row,col)
  const float* filt = (const float*)d_in[1];  // [256, 64, 64] f32
  float*       out  = (float*)d_out;          // [512, 512] f32
  const size_t HW = (size_t)HH * WW;

  int C = 1;
  if (ws_size >= 8 * HW * sizeof(float))      C = 8;
  else if (ws_size >= 4 * HW * sizeof(float)) C = 4;
  else if (ws_size >= 2 * HW * sizeof(float)) C = 2;

  if (C > 1) {
    float* partial = (float*)d_ws;
    splat_gather_accum<<<dim3(NTIL * C), dim3(256), 0, stream>>>(
        phw, filt, partial, NSPLAT, NSPLAT / C);
    reduce_partials<<<dim3((int)(HW / 256)), dim3(256), 0, stream>>>(partial, out, C);
  } else {
    splat_gather_accum<<<dim3(NTIL), dim3(256), 0, stream>>>(
        phw, filt, out, NSPLAT, NSPLAT);
  }
}